// GAE_84731114815725
// MI455X (gfx1250) — compile-verified
//
#include <hip/hip_runtime.h>
#include <hip/hip_bf16.h>

// ---------------------------------------------------------------------------
// GAE multi-view GCN autoencoder for MI455X (gfx1250, wave32, WMMA).
// Dense GEMMs use V_WMMA_F32_16X16X4_F32 (fp32 A/B, fp32 accum) since the
// workload is HBM-bound (~10 GB traffic vs ~55 GFLOP), so full-precision
// matrix ops cost nothing at the roofline. Graph aggregation uses
// global_atomic_add_f32 scatter.
// ---------------------------------------------------------------------------

typedef __attribute__((ext_vector_type(2))) float v2f;
typedef __attribute__((ext_vector_type(8))) float v8f;

#define NNODES 10000
#define NEDGES 320000

// ------------------------------ utility kernels ----------------------------

__global__ void zero_f32(float* __restrict__ p, long long n) {
  long long i = (long long)blockIdx.x * blockDim.x + threadIdx.x;
  if (i < n) p[i] = 0.0f;
}

__global__ void copy_f32(const float* __restrict__ s, float* __restrict__ d, long long n) {
  long long i = (long long)blockIdx.x * blockDim.x + threadIdx.x;
  if (i < n) d[i] = s[i];
}

__global__ void add_inplace_f32(float* __restrict__ out, const float* __restrict__ h, long long n) {
  long long i = (long long)blockIdx.x * blockDim.x + threadIdx.x;
  if (i < n) out[i] += h[i];
}

__global__ void degree_f32(const int* __restrict__ src, const int* __restrict__ dst,
                           float* __restrict__ dout, float* __restrict__ din, int E) {
  int e = blockIdx.x * blockDim.x + threadIdx.x;
  if (e < E) {
    atomicAdd(&dout[src[e]], 1.0f);
    atomicAdd(&din[dst[e]], 1.0f);
  }
}

// n = deg+extra > 0 ? rsqrt(max(deg+extra,1)) : 0   (extra=1 adds self-loops)
__global__ void norm_f32(const float* __restrict__ deg, float* __restrict__ nrm,
                         int n, float extra) {
  int i = blockIdx.x * blockDim.x + threadIdx.x;
  if (i < n) {
    float d = deg[i] + extra;
    nrm[i] = (d > 0.0f) ? rsqrtf(fmaxf(d, 1.0f)) : 0.0f;
  }
}

// out[dst[e], :] += h[src[e], :]   one thread per (edge, float4-group)
__global__ void scatter_add_f32(const int* __restrict__ src, const int* __restrict__ dst,
                                const float* __restrict__ h, float* __restrict__ out,
                                int E, int logF) {
  long long gid = (long long)blockIdx.x * blockDim.x + threadIdx.x;
  const int groups = 1 << (logF - 2);
  long long total = (long long)E * groups;
  if (gid >= total) return;
  int e  = (int)(gid >> (logF - 2));
  int f4 = ((int)(gid & (groups - 1))) << 2;
  const int F = 1 << logF;
  const float4 v = *(const float4*)(h + (long long)src[e] * F + f4);
  float* o = out + (long long)dst[e] * F + f4;
  atomicAdd(o + 0, v.x);
  atomicAdd(o + 1, v.y);
  atomicAdd(o + 2, v.z);
  atomicAdd(o + 3, v.w);
}

// x = act(x * nin[row] + b[col])
__global__ void bias_act_f32(float* __restrict__ x, const float* __restrict__ nin,
                             const float* __restrict__ b, int M, int logF, int relu) {
  long long gid = (long long)blockIdx.x * blockDim.x + threadIdx.x;
  long long total = (long long)M << logF;
  if (gid >= total) return;
  int i = (int)(gid >> logF);
  int f = (int)(gid & ((1 << logF) - 1));
  float v = x[gid] * nin[i] + b[f];
  if (relu) v = fmaxf(v, 0.0f);
  x[gid] = v;
}

// ------------------------------ WMMA GEMM ----------------------------------
// C[M,N] = epilogue(A[M,K] @ B[K,N]); one 16x16 tile per wave.
// Fragment layouts per CDNA5 ISA 7.12.2 (fp32, 16x16x4):
//   A: lane m=lane&15 holds row m; half-wave 0 -> K {k,k+1}, half-wave 1 -> K {k+2,k+3}
//   B: lane n=lane&15 holds col n; same K split
//   C: VGPR r -> row r (lanes 0-15) / r+8 (lanes 16-31), col = lane&15
__global__ void __launch_bounds__(256)
gemm_wmma_f32(const float* __restrict__ A, const float* __restrict__ B,
              float* __restrict__ C,
              const float* __restrict__ rowscale,   // nullable: * rowscale[row]
              const float* __restrict__ bias,       // nullable: + bias[col]
              int M, int N, int K, int relu, int acc) {
  const int lane = threadIdx.x & 31;
  const int wave = blockIdx.x * (blockDim.x >> 5) + (threadIdx.x >> 5);
  const int tilesN = N >> 4;
  const int tiles  = (M >> 4) * tilesN;
  if (wave >= tiles) return;                 // wave-uniform: EXEC stays all-ones
  const int tm = (wave / tilesN) << 4;
  const int tn = (wave % tilesN) << 4;
  const int mn = lane & 15;
  const int kk = (lane >> 4) << 1;           // 0 or 2
  const float* Arow = A + (long long)(tm + mn) * K;
  const float* Bcol = B + tn + mn;
  v8f c = {};
  for (int k = 0; k < K; k += 4) {
    v2f a, b;
    a[0] = Arow[k + kk];
    a[1] = Arow[k + kk + 1];
    b[0] = Bcol[(long long)(k + kk) * N];
    b[1] = Bcol[(long long)(k + kk + 1) * N];
    c = __builtin_amdgcn_wmma_f32_16x16x4_f32(false, a, false, b, (short)0, c, false, false);
  }
  const int rbase = (lane >> 4) << 3;        // +8 rows for upper half-wave
#pragma unroll
  for (int r = 0; r < 8; ++r) {
    const int row = tm + rbase + r;
    long long idx = (long long)row * N + tn + mn;
    float v = c[r];
    if (rowscale) v *= rowscale[row];
    if (bias)     v += bias[tn + mn];
    if (acc)      v += C[idx];
    if (relu)     v = fmaxf(v, 0.0f);
    C[idx] = v;
  }
}

// adj = sigmoid(T @ XH^T); writes the same tile to all 3 adjacency copies.
__global__ void __launch_bounds__(256)
decoder_wmma_f32(const float* __restrict__ T, const float* __restrict__ XH,
                 float* __restrict__ out, int Nn, int K) {
  const int lane = threadIdx.x & 31;
  const long long wave = (long long)blockIdx.x * (blockDim.x >> 5) + (threadIdx.x >> 5);
  const int tilesN = Nn >> 4;
  const long long tiles = (long long)tilesN * tilesN;
  if (wave >= tiles) return;                 // wave-uniform
  const int tm = (int)(wave / tilesN) << 4;
  const int tn = (int)(wave % tilesN) << 4;
  const int mn = lane & 15;
  const int kk = (lane >> 4) << 1;
  const float* Arow = T  + (long long)(tm + mn) * K;
  const float* Brow = XH + (long long)(tn + mn) * K;  // B[k][n] = XH[n][k]
  v8f c = {};
  for (int k = 0; k < K; k += 4) {
    v2f a, b;
    a[0] = Arow[k + kk];
    a[1] = Arow[k + kk + 1];
    b[0] = Brow[k + kk];
    b[1] = Brow[k + kk + 1];
    c = __builtin_amdgcn_wmma_f32_16x16x4_f32(false, a, false, b, (short)0, c, false, false);
  }
  const long long NN = (long long)Nn * Nn;
  const int rbase = (lane >> 4) << 3;
#pragma unroll
  for (int r = 0; r < 8; ++r) {
    long long idx = (long long)(tm + rbase + r) * Nn + tn + mn;
    float v = c[r];
    v = 1.0f / (1.0f + __expf(-v));
    out[idx]          = v;
    out[idx + NN]     = v;
    out[idx + 2 * NN] = v;
  }
}

// ------------------------------ host driver --------------------------------

extern "C" void kernel_launch(void* const* d_in, const int* in_sizes, int n_in,
                              void* d_out, int out_size, void* d_ws, size_t ws_size,
                              hipStream_t stream) {
  (void)in_sizes; (void)n_in; (void)out_size; (void)ws_size;
  const int N = NNODES, E = NEDGES;

  const float* feat[3] = {(const float*)d_in[0], (const float*)d_in[1], (const float*)d_in[2]};
  const int *gsrc[4], *gdst[4];
  for (int g = 0; g < 4; ++g) {
    gsrc[g] = (const int*)d_in[3 + 2 * g];
    gdst[g] = (const int*)d_in[4 + 2 * g];
  }
  const float *w0[3], *b0[3], *w1[3], *b1[3];
  for (int v = 0; v < 3; ++v) {
    w0[v] = (const float*)d_in[11 + 4 * v];
    b0[v] = (const float*)d_in[12 + 4 * v];
    w1[v] = (const float*)d_in[13 + 4 * v];
    b1[v] = (const float*)d_in[14 + 4 * v];
  }
  const float* fw[3] = {(const float*)d_in[23], (const float*)d_in[24], (const float*)d_in[25]};
  const float* fcw  = (const float*)d_in[26];
  const float* fcb  = (const float*)d_in[27];
  const float* wm0  = (const float*)d_in[28];
  const float* bm0  = (const float*)d_in[29];
  const float* wm1  = (const float*)d_in[30];
  const float* bm1  = (const float*)d_in[31];
  const float* decw = (const float*)d_in[32];
  float* out = (float*)d_out;

  // workspace carve-up (floats); total ~22M floats (~88 MB)
  float* ws      = (float*)d_ws;
  float* deg_out = ws;                                  // 4N
  float* deg_in  = deg_out + 4LL * N;                   // 4N
  float* nrm_out = deg_in  + 4LL * N;                   // 4N
  float* nrm_in  = nrm_out + 4LL * N;                   // 4N
  float* bufA    = nrm_in  + 4LL * N;                   // N*512
  float* bufB    = bufA + (long long)N * 512;           // N*512
  float* h0      = bufB + (long long)N * 512;           // N*256
  float* h1      = h0 + (long long)N * 256;             // N*256
  float* h2      = h1 + (long long)N * 256;             // N*256
  float* yb      = h2 + (long long)N * 256;             // N*256
  float* tb      = yb + (long long)N * 256;             // N*128
  float* hv[3]   = {h0, h1, h2};

  auto nb = [](long long n, int t) { return (int)((n + t - 1) / t); };
  auto gemm = [&](const float* A, const float* B, float* C, const float* rs,
                  const float* bias, int M, int Nc, int K, int relu, int acc) {
    const int wpb = 8;                                  // 8 waves / 256-thread block
    int tiles = (M >> 4) * (Nc >> 4);
    gemm_wmma_f32<<<(tiles + wpb - 1) / wpb, wpb * 32, 0, stream>>>(
        A, B, C, rs, bias, M, Nc, K, relu, acc);
  };

  // ---- degrees & norms for the 4 graphs (graph 3 gets +1 self-loop) ----
  zero_f32<<<nb(8LL * N, 256), 256, 0, stream>>>(deg_out, 8LL * N);
  for (int g = 0; g < 4; ++g)
    degree_f32<<<nb(E, 256), 256, 0, stream>>>(gsrc[g], gdst[g], deg_out + g * N, deg_in + g * N, E);
  for (int g = 0; g < 4; ++g) {
    float extra = (g == 3) ? 1.0f : 0.0f;
    norm_f32<<<nb(N, 256), 256, 0, stream>>>(deg_out + g * N, nrm_out + g * N, N, extra);
    norm_f32<<<nb(N, 256), 256, 0, stream>>>(deg_in  + g * N, nrm_in  + g * N, N, extra);
  }

  // ---- per-view 2-layer GCN stacks ----
  for (int v = 0; v < 3; ++v) {
    // layer 0: 512 -> 512, relu
    gemm(feat[v], w0[v], bufA, nrm_out + v * N, nullptr, N, 512, 512, 0, 0);
    zero_f32<<<nb((long long)N * 512, 256), 256, 0, stream>>>(bufB, (long long)N * 512);
    scatter_add_f32<<<nb((long long)E * 128, 256), 256, 0, stream>>>(gsrc[v], gdst[v], bufA, bufB, E, 9);
    bias_act_f32<<<nb((long long)N * 512, 256), 256, 0, stream>>>(bufB, nrm_in + v * N, b0[v], N, 9, 1);
    // layer 1: 512 -> 256, identity
    gemm(bufB, w1[v], bufA, nrm_out + v * N, nullptr, N, 256, 512, 0, 0);
    zero_f32<<<nb((long long)N * 256, 256), 256, 0, stream>>>(hv[v], (long long)N * 256);
    scatter_add_f32<<<nb((long long)E * 64, 256), 256, 0, stream>>>(gsrc[v], gdst[v], bufA, hv[v], E, 8);
    bias_act_f32<<<nb((long long)N * 256, 256), 256, 0, stream>>>(hv[v], nrm_in + v * N, b1[v], N, 8, 0);
  }

  // ---- FeatureFusion: y = h0@fw1 + h1@fw2 + h2@fw3 ; xh = y@fcw + fcb ----
  gemm(h0, fw[0], yb, nullptr, nullptr, N, 256, 256, 0, 0);
  gemm(h1, fw[1], yb, nullptr, nullptr, N, 256, 256, 0, 1);
  gemm(h2, fw[2], yb, nullptr, nullptr, N, 256, 256, 0, 1);
  gemm(yb, fcw, bufA, nullptr, fcb, N, 256, 256, 0, 0);   // xh1 -> bufA [N,256]

  // ---- fusion-graph GCN (self-loops folded into norms + explicit agg+=h) ----
  // layer 0: 256 -> 256, relu
  gemm(bufA, wm0, bufB, nrm_out + 3 * N, nullptr, N, 256, 256, 0, 0);
  zero_f32<<<nb((long long)N * 256, 256), 256, 0, stream>>>(yb, (long long)N * 256);
  scatter_add_f32<<<nb((long long)E * 64, 256), 256, 0, stream>>>(gsrc[3], gdst[3], bufB, yb, E, 8);
  add_inplace_f32<<<nb((long long)N * 256, 256), 256, 0, stream>>>(yb, bufB, (long long)N * 256);
  bias_act_f32<<<nb((long long)N * 256, 256), 256, 0, stream>>>(yb, nrm_in + 3 * N, bm0, N, 8, 1);
  // layer 1: 256 -> 128, identity
  gemm(yb, wm1, bufA, nrm_out + 3 * N, nullptr, N, 128, 256, 0, 0);
  zero_f32<<<nb((long long)N * 128, 256), 256, 0, stream>>>(bufB, (long long)N * 128);
  scatter_add_f32<<<nb((long long)E * 32, 256), 256, 0, stream>>>(gsrc[3], gdst[3], bufA, bufB, E, 7);
  add_inplace_f32<<<nb((long long)N * 128, 256), 256, 0, stream>>>(bufB, bufA, (long long)N * 128);
  bias_act_f32<<<nb((long long)N * 128, 256), 256, 0, stream>>>(bufB, nrm_in + 3 * N, bm1, N, 7, 0);
  // bufB now holds xh [N,128]

  // ---- InnerProductDecoder: adj = sigmoid((xh@decw) @ xh^T), 3 copies + xh ----
  gemm(bufB, decw, tb, nullptr, nullptr, N, 128, 128, 0, 0);
  {
    const int wpb = 8;
    long long tiles = (long long)(N / 16) * (N / 16);   // 625*625
    decoder_wmma_f32<<<(int)((tiles + wpb - 1) / wpb), wpb * 32, 0, stream>>>(tb, bufB, out, N, 128);
  }
  copy_f32<<<nb((long long)N * 128, 256), 256, 0, stream>>>(bufB, out + 3LL * N * N, (long long)N * 128);
}